// SmoothGCN_43602507989840
// MI455X (gfx1250) — compile-verified
//
#include <hip/hip_runtime.h>

#define N_NODES  50000
#define N_EDGES  1600000
#define IN_FEAT  256
#define HID      128
#define OUT_FEAT 256

typedef __attribute__((ext_vector_type(2))) float v2f;
typedef __attribute__((ext_vector_type(8))) float v8f;

// ---------------------------------------------------------------------------
// Kernel 1: for a 16-row tile of nodes, compute simultaneously
//   h[r, :]  = x @ W_node + (b_node + b_edge)        [HID wide]
//   xe[r, :] = x @ W_edge                            [HID wide]
// Block = 256 threads = 8 waves; wave w owns N-tile w (16 cols of HID=128).
// K loop: IN_FEAT=256 in steps of 4 using v_wmma_f32_16x16x4_f32.
// A-operand (x tile) loads are shared between the two accumulators.
// ---------------------------------------------------------------------------
__global__ __launch_bounds__(256)
void gemm_dual(const float* __restrict__ x,
               const float* __restrict__ Wn, const float* __restrict__ bn,
               const float* __restrict__ We, const float* __restrict__ be,
               float* __restrict__ h, float* __restrict__ xe)
{
    const int wave = threadIdx.x >> 5;
    const int lane = threadIdx.x & 31;
    const int half = lane >> 4;      // 0: lanes 0-15, 1: lanes 16-31
    const int l16  = lane & 15;
    const int row0 = blockIdx.x * 16;
    const int ncol = wave * 16 + l16;

    // fp32 A 16x4 layout: lane m (m=l16) holds K = 2*half + {0,1}
    const float* arow = x + (size_t)(row0 + l16) * IN_FEAT + 2 * half;

    v8f accN = {};
    v8f accE = {};
    for (int k = 0; k < IN_FEAT; k += 4) {
        v2f a = *(const v2f*)(arow + k);
        const int krow = k + 2 * half;       // B 4x16: lane n holds K = 2*half + {0,1}
        v2f bN, bE;
        bN.x = Wn[(size_t)(krow    ) * HID + ncol];
        bN.y = Wn[(size_t)(krow + 1) * HID + ncol];
        bE.x = We[(size_t)(krow    ) * HID + ncol];
        bE.y = We[(size_t)(krow + 1) * HID + ncol];
        accN = __builtin_amdgcn_wmma_f32_16x16x4_f32(
            false, a, false, bN, (short)0, accN, false, false);
        accE = __builtin_amdgcn_wmma_f32_16x16x4_f32(
            false, a, false, bE, (short)0, accE, false, false);
    }

    const float bias = bn[ncol] + be[ncol];
    #pragma unroll
    for (int r = 0; r < 8; ++r) {
        // C/D layout: VGPR r = row (r + 8*half), col = ncol
        const size_t off = (size_t)(row0 + r + 8 * half) * HID + ncol;
        h[off]  = accN[r] + bias;
        xe[off] = accE[r];
    }
}

// ---------------------------------------------------------------------------
// Kernel 2: h[dst] += w_e * xe[src]  (scatter in the HID=128 projected space)
// One wave per edge; each lane owns 4 consecutive features (float4 gather,
// 4 no-return global f32 atomics). xe (25.6 MB) is L2-resident on MI455X.
// ---------------------------------------------------------------------------
__global__ __launch_bounds__(256)
void edge_scatter(const float* __restrict__ xe, const float* __restrict__ w,
                  const int* __restrict__ src, const int* __restrict__ dst,
                  float* __restrict__ h)
{
    const int e = blockIdx.x * 8 + (threadIdx.x >> 5);
    if (e >= N_EDGES) return;
    const int lane = threadIdx.x & 31;

    const int   s  = src[e];
    const int   d  = dst[e];
    const float we = w[e];

    const float4 v = *(const float4*)(xe + (size_t)s * HID + lane * 4);
    float* out = h + (size_t)d * HID + lane * 4;
    unsafeAtomicAdd(out + 0, v.x * we);
    unsafeAtomicAdd(out + 1, v.y * we);
    unsafeAtomicAdd(out + 2, v.z * we);
    unsafeAtomicAdd(out + 3, v.w * we);
}

// ---------------------------------------------------------------------------
// Kernel 3: out = h @ W_mlp + b_mlp   (50000x128 @ 128x256)
// Block = 256 threads = 8 waves; wave w owns N-tiles {2w, 2w+1} of OUT=256,
// sharing A-operand loads between the two accumulators. K=128 -> 32 steps.
// ---------------------------------------------------------------------------
__global__ __launch_bounds__(256)
void gemm_mlp(const float* __restrict__ h, const float* __restrict__ Wm,
              const float* __restrict__ bm, float* __restrict__ out)
{
    const int wave = threadIdx.x >> 5;
    const int lane = threadIdx.x & 31;
    const int half = lane >> 4;
    const int l16  = lane & 15;
    const int row0 = blockIdx.x * 16;
    const int n0   = wave * 32 + l16;     // tiles at n0 and n0+16

    const float* arow = h + (size_t)(row0 + l16) * HID + 2 * half;

    v8f acc0 = {};
    v8f acc1 = {};
    for (int k = 0; k < HID; k += 4) {
        v2f a = *(const v2f*)(arow + k);
        const int krow = k + 2 * half;
        v2f b0, b1;
        b0.x = Wm[(size_t)(krow    ) * OUT_FEAT + n0];
        b0.y = Wm[(size_t)(krow + 1) * OUT_FEAT + n0];
        b1.x = Wm[(size_t)(krow    ) * OUT_FEAT + n0 + 16];
        b1.y = Wm[(size_t)(krow + 1) * OUT_FEAT + n0 + 16];
        acc0 = __builtin_amdgcn_wmma_f32_16x16x4_f32(
            false, a, false, b0, (short)0, acc0, false, false);
        acc1 = __builtin_amdgcn_wmma_f32_16x16x4_f32(
            false, a, false, b1, (short)0, acc1, false, false);
    }

    const float bias0 = bm[n0];
    const float bias1 = bm[n0 + 16];
    #pragma unroll
    for (int r = 0; r < 8; ++r) {
        const size_t row = (size_t)(row0 + r + 8 * half);
        out[row * OUT_FEAT + n0     ] = acc0[r] + bias0;
        out[row * OUT_FEAT + n0 + 16] = acc1[r] + bias1;
    }
}

// ---------------------------------------------------------------------------
extern "C" void kernel_launch(void* const* d_in, const int* in_sizes, int n_in,
                              void* d_out, int out_size, void* d_ws, size_t ws_size,
                              hipStream_t stream)
{
    (void)in_sizes; (void)n_in; (void)out_size; (void)ws_size;

    const float* x   = (const float*)d_in[0];
    const float* w   = (const float*)d_in[1];
    const int*   src = (const int*)  d_in[2];
    const int*   dst = (const int*)  d_in[3];
    const float* Wn  = (const float*)d_in[4];
    const float* bn  = (const float*)d_in[5];
    const float* We  = (const float*)d_in[6];
    const float* be  = (const float*)d_in[7];
    const float* Wm  = (const float*)d_in[8];
    const float* bm  = (const float*)d_in[9];
    float* out = (float*)d_out;

    float* h  = (float*)d_ws;                      // [N_NODES, HID]
    float* xe = h + (size_t)N_NODES * HID;         // [N_NODES, HID]

    // 1) h = x@W_node + b_node + b_edge ; xe = x@W_edge   (WMMA, fused)
    gemm_dual<<<N_NODES / 16, 256, 0, stream>>>(x, Wn, bn, We, be, h, xe);

    // 2) h[dst] += w_e * xe[src]  (projected-space scatter, L2 atomics)
    edge_scatter<<<N_EDGES / 8, 256, 0, stream>>>(xe, w, src, dst, h);

    // 3) out = h@W_mlp + b_mlp   (WMMA)
    gemm_mlp<<<N_NODES / 16, 256, 0, stream>>>(h, Wm, bm, out);
}